// DilationDense_5609227288817
// MI455X (gfx1250) — compile-verified
//
#include <hip/hip_runtime.h>
#include <stdint.h>

// Tropical (max-plus) dense: out[b,u] = max(bias[u], max_i (x[b,i] + k[i,u]))
// B=16384, I=128, U=128, fp32.
//
// gfx1250 paths used: TENSOR_LOAD_TO_LDS (TDM DMA of the 64KB kernel matrix
// into LDS), s_wait_tensorcnt, global_prefetch_b8, ds_load_b128 from LDS.

#define B_TOTAL 16384
#define I_DIM   128
#define U_DIM   128
#define BM      64      // x rows per block
#define TM      4       // b rows per thread
#define TN      8       // u cols per thread
#define NTHREADS 256

typedef unsigned int u32x4 __attribute__((ext_vector_type(4)));
typedef int          i32x4 __attribute__((ext_vector_type(4)));
typedef int          i32x8 __attribute__((ext_vector_type(8)));

__global__ __launch_bounds__(NTHREADS)
void tropical_dense_kernel(const float* __restrict__ x,
                           const float* __restrict__ kmat,
                           const float* __restrict__ bias,
                           float* __restrict__ out)
{
    __shared__ float lds_k[I_DIM * U_DIM];   // 64 KB, whole kernel matrix

    const int t  = threadIdx.x;
    const int u0 = (t & 15) * TN;                         // 0..120 step 8
    const int b0 = blockIdx.x * BM + (t >> 4) * TM;       // 4 consecutive rows

    // Prefetch the start of this thread's x rows while the TDM runs.
    __builtin_prefetch(x + (size_t)b0 * I_DIM, 0, 0);

    // ---- Stage kernel matrix into LDS -------------------------------------
#if __has_builtin(__builtin_amdgcn_tensor_load_to_lds) && \
    __has_builtin(__builtin_amdgcn_s_wait_tensorcnt)
    if (t < 32) {   // wave 0 only (wave-uniform branch); TDM is a per-wave DMA
        const unsigned long long ga = (unsigned long long)(uintptr_t)kmat;
        const unsigned int lds_off  =
            (unsigned int)(unsigned long long)(uintptr_t)&lds_k[0];

        // D# group 0 (128b): count=1 | lds_addr | global_addr[56:0] | type=2
        u32x4 g0;
        g0.x = 1u;                                   // count=1, no gather
        g0.y = lds_off;                              // LDS byte address
        g0.z = (unsigned int)(ga & 0xffffffffull);   // global_addr[31:0]
        g0.w = (unsigned int)((ga >> 32) & 0x1ffffffull) | (2u << 30); // +type=2

        // D# group 1 (256b): data_size=4B; tensor 128x128; tile 128x128;
        // dim0 stride = 128 elements; no multicast / barrier / pad / iterate.
        i32x8 g1;
        g1.s0 = (int)(2u << 16);          // data_size = 2 (4 bytes)
        g1.s1 = (int)(128u << 16);        // tensor_dim0[15:0] = 128
        g1.s2 = (int)(128u << 16);        // tensor_dim1[15:0] = 128
        g1.s3 = (int)(128u << 16);        // tile_dim0 = 128
        g1.s4 = 128;                      // tile_dim1 = 128, tile_dim2 = 0
        g1.s5 = 128;                      // tensor_dim0_stride = 128
        g1.s6 = 0;                        // dim0_stride hi, dim1_stride lo
        g1.s7 = 0;

        i32x4 g2 = {0, 0, 0, 0};          // 2-D tensor: groups 2/3 unused
        i32x4 g3 = {0, 0, 0, 0};

#if __clang_major__ >= 23
        i32x8 g4 = {0, 0, 0, 0, 0, 0, 0, 0};
        __builtin_amdgcn_tensor_load_to_lds(g0, g1, g2, g3, g4, 0);
#else
        __builtin_amdgcn_tensor_load_to_lds(g0, g1, g2, g3, 0);
#endif
        __builtin_amdgcn_s_wait_tensorcnt(0);   // s_wait_tensorcnt 0
    }
#else
    // Fallback: cooperative float4 copy
    {
        const float4* src = reinterpret_cast<const float4*>(kmat);
        float4*       dst = reinterpret_cast<float4*>(lds_k);
        for (int idx = t; idx < (I_DIM * U_DIM) / 4; idx += NTHREADS)
            dst[idx] = src[idx];
    }
#endif
    __syncthreads();

    // ---- Register-blocked max-plus ----------------------------------------
    const float NEG_INF = -__builtin_huge_valf();
    float acc[TM][TN];
#pragma unroll
    for (int j = 0; j < TM; ++j)
#pragma unroll
        for (int u = 0; u < TN; ++u)
            acc[j][u] = NEG_INF;

    const float4* x4 = reinterpret_cast<const float4*>(x);

#pragma unroll 1
    for (int ic = 0; ic < I_DIM; ic += 4) {
        float4 xv[TM];
#pragma unroll
        for (int j = 0; j < TM; ++j)
            xv[j] = x4[(size_t)(b0 + j) * (I_DIM / 4) + (ic >> 2)];

#pragma unroll
        for (int ii = 0; ii < 4; ++ii) {
            const float4 ka =
                *reinterpret_cast<const float4*>(&lds_k[(ic + ii) * U_DIM + u0]);
            const float4 kb =
                *reinterpret_cast<const float4*>(&lds_k[(ic + ii) * U_DIM + u0 + 4]);
            const float kv[TN] = {ka.x, ka.y, ka.z, ka.w,
                                  kb.x, kb.y, kb.z, kb.w};
#pragma unroll
            for (int j = 0; j < TM; ++j) {
                const float xs = (ii == 0) ? xv[j].x :
                                 (ii == 1) ? xv[j].y :
                                 (ii == 2) ? xv[j].z : xv[j].w;
#pragma unroll
                for (int u = 0; u < TN; ++u)
                    acc[j][u] = fmaxf(acc[j][u], xs + kv[u]);
            }
        }
    }

    // ---- Bias max + store --------------------------------------------------
    float bv[TN];
#pragma unroll
    for (int u = 0; u < TN; ++u)
        bv[u] = bias[u0 + u];

#pragma unroll
    for (int j = 0; j < TM; ++j) {
        float4 o0, o1;
        o0.x = fmaxf(acc[j][0], bv[0]);
        o0.y = fmaxf(acc[j][1], bv[1]);
        o0.z = fmaxf(acc[j][2], bv[2]);
        o0.w = fmaxf(acc[j][3], bv[3]);
        o1.x = fmaxf(acc[j][4], bv[4]);
        o1.y = fmaxf(acc[j][5], bv[5]);
        o1.z = fmaxf(acc[j][6], bv[6]);
        o1.w = fmaxf(acc[j][7], bv[7]);
        float* orow = out + (size_t)(b0 + j) * U_DIM + u0;
        *reinterpret_cast<float4*>(orow)     = o0;
        *reinterpret_cast<float4*>(orow + 4) = o1;
    }
}

extern "C" void kernel_launch(void* const* d_in, const int* in_sizes, int n_in,
                              void* d_out, int out_size, void* d_ws, size_t ws_size,
                              hipStream_t stream)
{
    (void)n_in; (void)out_size; (void)d_ws; (void)ws_size;
    const float* x    = (const float*)d_in[0];
    const float* kmat = (const float*)d_in[1];
    const float* bias = (const float*)d_in[2];
    float*       out  = (float*)d_out;

    const int B = in_sizes[0] / I_DIM;   // 16384
    dim3 grid(B / BM);                   // 256 blocks
    dim3 block(NTHREADS);                // 8 wave32s
    tropical_dense_kernel<<<grid, block, 0, stream>>>(x, kmat, bias, out);
}